// PNA_65515431133326
// MI455X (gfx1250) — compile-verified
//
#include <hip/hip_runtime.h>
#include <math.h>

#define N_NODES  50000
#define N_EDGES  150000
#define N_GRAPHS 2000
#define D        75
#define E_D      50
#define TOWERS   5
#define F_OUT    15
#define N_LAYERS 4
#define KPAD     76      // K padded to 19*4
#define KSTEPS   19
#define NCOLS    750     // 2 * TOWERS * D (A-part | B-part)
#define NPAD     768     // 12 tiles of 64
#define AVG_DEG_LOG 1.1308269950712107f

typedef __attribute__((ext_vector_type(2))) float v2f;
typedef __attribute__((ext_vector_type(8))) float v8f;

// ---------------- weight packing (K-pair interleaved) ----------------
// Logical W[l][k][j], k in [0,76), j in [0,768): j<375 -> dst-part (Wpre rows 0..74),
// 375<=j<750 -> src-part (Wpre rows 75..149), zero padding elsewhere.
// Stored as Wpack[l][(k/2)*768 + j][k&1] so a B-fragment {W[k][j], W[k+1][j]}
// is one aligned 8-byte load (global_load_b64), coalesced across lanes.
__global__ void k_pack(const float* __restrict__ Wpre, float* __restrict__ Wpack) {
  int id = blockIdx.x * blockDim.x + threadIdx.x;
  const int total = N_LAYERS * KPAD * NPAD;
  if (id >= total) return;
  int l   = id / (KPAD * NPAD);
  int rem = id % (KPAD * NPAD);         // = (kp*NPAD + j)*2 + r
  int r   = rem & 1;
  int t2  = rem >> 1;
  int kp  = t2 / NPAD;
  int j   = t2 % NPAD;
  int k   = kp * 2 + r;
  float v = 0.0f;
  if (k < D && j < NCOLS) {
    int part = j / 375;                 // 0 = x[dst] rows, 1 = x[src] rows
    int jj = j % 375;
    int t = jj / D, f = jj % D;
    v = Wpre[(((size_t)(l * TOWERS + t) * (3 * D)) + part * D + k) * D + f];
  }
  Wpack[id] = v;
}

// etab[l][a][c] = edge_emb[a] @ We[l] + be[l]
__global__ void k_etab(const float* __restrict__ edge_emb, const float* __restrict__ We,
                       const float* __restrict__ be, float* __restrict__ etab) {
  int id = blockIdx.x * blockDim.x + threadIdx.x;
  if (id >= N_LAYERS * 4 * D) return;
  int l = id / (4 * D);
  int a = (id / D) % 4;
  int c = id % D;
  float s = be[l * D + c];
  for (int q = 0; q < E_D; ++q)
    s += edge_emb[a * E_D + q] * We[(size_t)(l * E_D + q) * D + c];
  etab[id] = s;
}

// Ctab[l][a][t*75+f] = etab[l][a] @ Wpre[l,t,150:225,:] + bpre[l,t]
__global__ void k_ctab(const float* __restrict__ etab, const float* __restrict__ Wpre,
                       const float* __restrict__ bpre, float* __restrict__ Ctab) {
  int id = blockIdx.x * blockDim.x + threadIdx.x;
  if (id >= N_LAYERS * 4 * TOWERS * D) return;
  int l = id / (4 * TOWERS * D);
  int a = (id / (TOWERS * D)) % 4;
  int j = id % (TOWERS * D);
  int t = j / D, f = j % D;
  float s = bpre[(l * TOWERS + t) * D + f];
  const float* et = etab + (l * 4 + a) * D;
  for (int c = 0; c < D; ++c)
    s += et[c] * Wpre[((size_t)(l * TOWERS + t) * (3 * D) + 2 * D + c) * D + f];
  Ctab[id] = s;
}

// ---------------- CSR build over dst ----------------
__global__ void k_zero_deg(int* __restrict__ deg) {
  int i = blockIdx.x * blockDim.x + threadIdx.x;
  if (i < N_NODES) deg[i] = 0;
}
__global__ void k_count(const int* __restrict__ dst, int* __restrict__ deg) {
  int e = blockIdx.x * blockDim.x + threadIdx.x;
  if (e < N_EDGES) atomicAdd(&deg[dst[e]], 1);
}
__global__ void k_scan(const int* __restrict__ deg, int* __restrict__ offsets,
                       int* __restrict__ cursor) {
  __shared__ int ps[1024];
  const int CH = 49;                          // 1024*49 >= 50000
  int t = threadIdx.x;
  int base = t * CH;
  int s = 0;
  for (int i = 0; i < CH; ++i) { int idx = base + i; if (idx < N_NODES) s += deg[idx]; }
  ps[t] = s;
  __syncthreads();
  for (int off = 1; off < 1024; off <<= 1) {
    int v = (t >= off) ? ps[t - off] : 0;
    __syncthreads();
    ps[t] += v;
    __syncthreads();
  }
  int run = (t > 0) ? ps[t - 1] : 0;
  for (int i = 0; i < CH; ++i) {
    int idx = base + i;
    if (idx < N_NODES) { offsets[idx] = run; cursor[idx] = run; run += deg[idx]; }
  }
  if (t == 1023) offsets[N_NODES] = ps[1023];
}
__global__ void k_fill(const int* __restrict__ dst, int* __restrict__ cursor,
                       int* __restrict__ csr) {
  int e = blockIdx.x * blockDim.x + threadIdx.x;
  if (e < N_EDGES) { int p = atomicAdd(&cursor[dst[e]], 1); csr[p] = e; }
}
__global__ void k_sortcsr(const int* __restrict__ offsets, int* __restrict__ csr) {
  int n = blockIdx.x * blockDim.x + threadIdx.x;
  if (n >= N_NODES) return;
  int lo = offsets[n], hi = offsets[n + 1];
  for (int i = lo + 1; i < hi; ++i) {
    int key = csr[i];
    int j = i - 1;
    while (j >= lo && csr[j] > key) { csr[j + 1] = csr[j]; --j; }
    csr[j + 1] = key;
  }
}

// ---------------- initial node embedding ----------------
__global__ void k_init_h(const int* __restrict__ x, const float* __restrict__ node_emb,
                         float* __restrict__ Hpad) {
  int n = blockIdx.x, f = threadIdx.x;
  if (f < D)       Hpad[(size_t)n * KPAD + f] = node_emb[x[n] * D + f];
  else if (f == D) Hpad[(size_t)n * KPAD + D] = 0.0f;
}

// ---------------- WMMA fp32 GEMM: AB[N,768] = Hpad[N,76] @ W[76,768] ----------------
// Block = 4 waves = 64 M-rows. X tile staged in LDS once (contiguous, coalesced),
// each wave keeps all 19 A-fragments in registers and sweeps every N-tile with them:
// X is read from global exactly once per layer. W is K-pair interleaved so every
// B-fragment is a single coalesced global_load_b64; panels stay L1/L2-resident.
__global__ void k_gemm_wmma(const float* __restrict__ X, const float* __restrict__ W,
                            float* __restrict__ Out, int M) {
  __shared__ float xs[64 * KPAD];             // 19.4 KB
  const int tid  = threadIdx.x;
  const int lane = tid & 31;
  const int wave = tid >> 5;
  const int mbase = blockIdx.x * 64;
  {
    int rows = M - mbase; if (rows > 64) rows = 64;
    const int span = rows * KPAD;             // X row stride == KPAD -> linear span
    const float* xsrc = X + (size_t)mbase * KPAD;
    for (int i = tid; i < span; i += 128) xs[i] = xsrc[i];
  }
  __syncthreads();
  const int m0 = mbase + wave * 16;
  if (m0 >= M) return;                        // wave-uniform: EXEC all-ones at WMMAs
  const int kb   = (lane >> 4) * 2;           // A/B fragment K selector per ISA layout
  const int lrow = wave * 16 + (lane & 15);
  v2f afrag[KSTEPS];
#pragma unroll
  for (int ks = 0; ks < KSTEPS; ++ks) {
    afrag[ks].x = xs[lrow * KPAD + ks * 4 + kb];
    afrag[ks].y = xs[lrow * KPAD + ks * 4 + kb + 1];
  }
  const int rbase = m0 + (lane >> 4) * 8;
  for (int nt = 0; nt < NPAD / 64; ++nt) {
    const int ncol = nt * 64 + (lane & 15);
    v8f acc0 = {}, acc1 = {}, acc2 = {}, acc3 = {};
#pragma unroll
    for (int ks = 0; ks < KSTEPS; ++ks) {
      // k = ks*4 + kb is even -> pair base (k/2) = ks*2 + kb/2
      const float* wbase = W + ((size_t)(ks * 2 + (kb >> 1)) * NPAD + ncol) * 2;
      v2f b0 = *(const v2f*)(wbase);          // {W[k][n+0],  W[k+1][n+0]}
      v2f b1 = *(const v2f*)(wbase + 32);     // {W[k][n+16], W[k+1][n+16]}
      v2f b2 = *(const v2f*)(wbase + 64);     // {W[k][n+32], W[k+1][n+32]}
      v2f b3 = *(const v2f*)(wbase + 96);     // {W[k][n+48], W[k+1][n+48]}
      acc0 = __builtin_amdgcn_wmma_f32_16x16x4_f32(false, afrag[ks], false, b0, (short)0, acc0, false, false);
      acc1 = __builtin_amdgcn_wmma_f32_16x16x4_f32(false, afrag[ks], false, b1, (short)0, acc1, false, false);
      acc2 = __builtin_amdgcn_wmma_f32_16x16x4_f32(false, afrag[ks], false, b2, (short)0, acc2, false, false);
      acc3 = __builtin_amdgcn_wmma_f32_16x16x4_f32(false, afrag[ks], false, b3, (short)0, acc3, false, false);
    }
#pragma unroll
    for (int v = 0; v < 8; ++v) {
      float* op = Out + (size_t)(rbase + v) * NPAD + ncol;
      op[0]  = acc0[v];
      op[16] = acc1[v];
      op[32] = acc2[v];
      op[48] = acc3[v];
    }
  }
}

// ---------------- fused aggregation + post-NN + Wlin (one block per node) ----------------
__global__ void k_agg_post(int l, const float* __restrict__ AB, const float* __restrict__ Hpad,
                           const float* __restrict__ Ctab, const int* __restrict__ offsets,
                           const int* __restrict__ csr, const int* __restrict__ src,
                           const int* __restrict__ attr,
                           const float* __restrict__ Wpost, const float* __restrict__ bpost,
                           const float* __restrict__ Wlin, const float* __restrict__ blin,
                           float* __restrict__ Hlin) {
  __shared__ float sMean[TOWERS * D], sMin[TOWERS * D], sMax[TOWERS * D], sStd[TOWERS * D];
  __shared__ float xrow[D], outv[D];
  __shared__ float s_amp, s_inv;
  const int n = blockIdx.x, tid = threadIdx.x;
  const int lo = offsets[n], hi = offsets[n + 1];
  const int degi = hi - lo;
  if (tid < D) xrow[tid] = Hpad[(size_t)n * KPAD + tid];
  if (tid == 0) {
    float dm = fmaxf((float)degi, 1.0f);
    float amp = logf(dm + 1.0f) / AVG_DEG_LOG;
    s_amp = amp; s_inv = 1.0f / amp;
  }
  const float* Ct = Ctab + (size_t)l * 4 * (TOWERS * D);
  for (int j = tid; j < TOWERS * D; j += blockDim.x) {
    float a = AB[(size_t)n * NPAD + j];
    float s = 0.f, q = 0.f, mn = 3.0e38f, mx = -3.0e38f;
    for (int e = lo; e < hi; ++e) {
      int ed = csr[e];
      float mv = a + AB[(size_t)src[ed] * NPAD + TOWERS * D + j] + Ct[attr[ed] * (TOWERS * D) + j];
      s += mv; q += mv * mv;
      mn = fminf(mn, mv); mx = fmaxf(mx, mv);
    }
    float denom = fmaxf((float)degi, 1.0f);
    float mean = s / denom;
    float msq  = q / denom;
    sMean[j] = mean;
    sStd[j]  = sqrtf(fmaxf(msq - mean * mean, 0.0f) + 1e-5f);
    sMin[j]  = (degi > 0) ? mn : 0.0f;
    sMax[j]  = (degi > 0) ? mx : 0.0f;
  }
  __syncthreads();
  const float amp = s_amp, inv = s_inv;
  if (tid < D) {
    const int t = tid / F_OUT, fo = tid % F_OUT;
    const float* Wp = Wpost + (size_t)((l * TOWERS + t) * (13 * D)) * F_OUT + fo;
    float acc = bpost[(l * TOWERS + t) * F_OUT + fo];
    for (int c = 0; c < D; ++c) acc += xrow[c] * Wp[c * F_OUT];
    const float* bases[4] = { sMean, sMin, sMax, sStd };
    for (int k = 0; k < 4; ++k) {
      const float* S   = bases[k] + t * D;
      const float* W1p = Wp + (size_t)(1 * D + k * D) * F_OUT;   // identity scaler rows
      const float* W2p = Wp + (size_t)(5 * D + k * D) * F_OUT;   // amplification rows
      const float* W3p = Wp + (size_t)(9 * D + k * D) * F_OUT;   // attenuation rows
      for (int f = 0; f < D; ++f)
        acc += S[f] * (W1p[f * F_OUT] + amp * W2p[f * F_OUT] + inv * W3p[f * F_OUT]);
    }
    outv[tid] = acc;
  }
  __syncthreads();
  if (tid < D) {
    float r = blin[l * D + tid];
    for (int c = 0; c < D; ++c) r += outv[c] * Wlin[(size_t)(l * D + c) * D + tid];
    Hlin[(size_t)n * D + tid] = r;
  }
}

// ---------------- batchnorm (training-mode stats) + relu ----------------
__global__ void k_bn_zero(float* __restrict__ bnsum, float* __restrict__ bnsq) {
  int t = threadIdx.x;
  if (t < D) { bnsum[t] = 0.f; bnsq[t] = 0.f; }
}
__global__ void k_bn_reduce(const float* __restrict__ Hlin, float* __restrict__ bnsum,
                            float* __restrict__ bnsq) {
  int f = threadIdx.x;
  if (f >= D) return;
  int rows_per = (N_NODES + gridDim.x - 1) / gridDim.x;
  int r0 = blockIdx.x * rows_per;
  int r1 = r0 + rows_per; if (r1 > N_NODES) r1 = N_NODES;
  float s = 0.f, q = 0.f;
  for (int n = r0; n < r1; ++n) { float v = Hlin[(size_t)n * D + f]; s += v; q += v * v; }
  atomicAdd(&bnsum[f], s);
  atomicAdd(&bnsq[f], q);
}
__global__ void k_bn_apply(int l, const float* __restrict__ Hlin, const float* __restrict__ bnsum,
                           const float* __restrict__ bnsq, const float* __restrict__ gamma,
                           const float* __restrict__ beta, float* __restrict__ Hpad) {
  int n = blockIdx.x, f = threadIdx.x;
  if (f == D) { Hpad[(size_t)n * KPAD + D] = 0.f; return; }
  if (f > D) return;
  float mu  = bnsum[f] * (1.0f / N_NODES);
  float var = bnsq[f] * (1.0f / N_NODES) - mu * mu;
  float v = Hlin[(size_t)n * D + f];
  float o = gamma[l * D + f] * (v - mu) * rsqrtf(var + 1e-5f) + beta[l * D + f];
  Hpad[(size_t)n * KPAD + f] = fmaxf(o, 0.f);
}

// ---------------- global add pool + readout MLP (batch is sorted) ----------------
__global__ void k_pool(const float* __restrict__ Hpad, const int* __restrict__ batch,
                       const float* __restrict__ W1, const float* __restrict__ b1,
                       const float* __restrict__ W2, const float* __restrict__ b2,
                       const float* __restrict__ W3, const float* __restrict__ b3,
                       float* __restrict__ out) {
  __shared__ float pooled[D], z1[50], z2[25];
  __shared__ int slo, shi;
  const int g = blockIdx.x, tid = threadIdx.x;
  if (tid == 0) {
    int lo = 0, hi = N_NODES;
    while (lo < hi) { int m = (lo + hi) >> 1; if (batch[m] < g) lo = m + 1; else hi = m; }
    slo = lo;
    int lo2 = lo, hi2 = N_NODES;
    while (lo2 < hi2) { int m = (lo2 + hi2) >> 1; if (batch[m] < g + 1) lo2 = m + 1; else hi2 = m; }
    shi = lo2;
  }
  __syncthreads();
  if (tid < D) {
    float s = 0.f;
    for (int n = slo; n < shi; ++n) s += Hpad[(size_t)n * KPAD + tid];
    pooled[tid] = s;
  }
  __syncthreads();
  if (tid < 50) {
    float z = b1[tid];
    for (int c = 0; c < D; ++c) z += pooled[c] * W1[c * 50 + tid];
    z1[tid] = fmaxf(z, 0.f);
  }
  __syncthreads();
  if (tid < 25) {
    float z = b2[tid];
    for (int c = 0; c < 50; ++c) z += z1[c] * W2[c * 25 + tid];
    z2[tid] = fmaxf(z, 0.f);
  }
  __syncthreads();
  if (tid == 0) {
    float z = b3[0];
    for (int c = 0; c < 25; ++c) z += z2[c] * W3[c];
    out[g] = z;
  }
}

extern "C" void kernel_launch(void* const* d_in, const int* in_sizes, int n_in,
                              void* d_out, int out_size, void* d_ws, size_t ws_size,
                              hipStream_t stream) {
  const int*   x        = (const int*)d_in[0];
  const int*   ei       = (const int*)d_in[1];
  const int*   src      = ei;
  const int*   dst      = ei + N_EDGES;
  const int*   attr     = (const int*)d_in[2];
  const int*   batch    = (const int*)d_in[3];
  const float* node_emb = (const float*)d_in[4];
  const float* edge_emb = (const float*)d_in[5];
  const float* We       = (const float*)d_in[6];
  const float* be       = (const float*)d_in[7];
  const float* Wpre     = (const float*)d_in[8];
  const float* bpre     = (const float*)d_in[9];
  const float* Wpost    = (const float*)d_in[10];
  const float* bpost    = (const float*)d_in[11];
  const float* Wlin     = (const float*)d_in[12];
  const float* blin     = (const float*)d_in[13];
  const float* gamma    = (const float*)d_in[14];
  const float* beta     = (const float*)d_in[15];
  const float* W1 = (const float*)d_in[16]; const float* b1 = (const float*)d_in[17];
  const float* W2 = (const float*)d_in[18]; const float* b2 = (const float*)d_in[19];
  const float* W3 = (const float*)d_in[20]; const float* b3 = (const float*)d_in[21];

  char* p = (char*)d_ws;
  auto alloc = [&](size_t bytes) -> void* {
    void* r = (void*)p;
    p += (bytes + 255) & ~(size_t)255;
    return r;
  };
  float* Wpack   = (float*)alloc(sizeof(float) * (size_t)N_LAYERS * KPAD * NPAD);
  float* etab    = (float*)alloc(sizeof(float) * N_LAYERS * 4 * D);
  float* Ctab    = (float*)alloc(sizeof(float) * N_LAYERS * 4 * TOWERS * D);
  int*   deg     = (int*)alloc(sizeof(int) * N_NODES);
  int*   offsets = (int*)alloc(sizeof(int) * (N_NODES + 1));
  int*   cursor  = (int*)alloc(sizeof(int) * N_NODES);
  int*   csr     = (int*)alloc(sizeof(int) * N_EDGES);
  float* bnsum   = (float*)alloc(sizeof(float) * D);
  float* bnsq    = (float*)alloc(sizeof(float) * D);
  float* Hpad    = (float*)alloc(sizeof(float) * (size_t)N_NODES * KPAD);
  float* Hlin    = (float*)alloc(sizeof(float) * (size_t)N_NODES * D);
  float* AB      = (float*)alloc(sizeof(float) * (size_t)N_NODES * NPAD);

  // ---- deterministic per-call preprocessing ----
  k_pack<<<(N_LAYERS * KPAD * NPAD + 255) / 256, 256, 0, stream>>>(Wpre, Wpack);
  k_etab<<<(N_LAYERS * 4 * D + 127) / 128, 128, 0, stream>>>(edge_emb, We, be, etab);
  k_ctab<<<(N_LAYERS * 4 * TOWERS * D + 127) / 128, 128, 0, stream>>>(etab, Wpre, bpre, Ctab);
  k_zero_deg<<<(N_NODES + 255) / 256, 256, 0, stream>>>(deg);
  k_count<<<(N_EDGES + 255) / 256, 256, 0, stream>>>(dst, deg);
  k_scan<<<1, 1024, 0, stream>>>(deg, offsets, cursor);
  k_fill<<<(N_EDGES + 255) / 256, 256, 0, stream>>>(dst, cursor, csr);
  k_sortcsr<<<(N_NODES + 127) / 128, 128, 0, stream>>>(offsets, csr);
  k_init_h<<<N_NODES, 96, 0, stream>>>(x, node_emb, Hpad);

  // ---- 4 PNA layers ----
  for (int l = 0; l < N_LAYERS; ++l) {
    k_gemm_wmma<<<(N_NODES + 63) / 64, 128, 0, stream>>>(
        Hpad, Wpack + (size_t)l * KPAD * NPAD, AB, N_NODES);
    k_agg_post<<<N_NODES, 128, 0, stream>>>(l, AB, Hpad, Ctab, offsets, csr, src, attr,
                                            Wpost, bpost, Wlin, blin, Hlin);
    k_bn_zero<<<1, 96, 0, stream>>>(bnsum, bnsq);
    k_bn_reduce<<<200, 96, 0, stream>>>(Hlin, bnsum, bnsq);
    k_bn_apply<<<N_NODES, 96, 0, stream>>>(l, Hlin, bnsum, bnsq, gamma, beta, Hpad);
  }

  // ---- pool + MLP readout ----
  k_pool<<<N_GRAPHS, 128, 0, stream>>>(Hpad, batch, W1, b1, W2, b2, W3, b3, (float*)d_out);
}